// MSEObserver_1726576855998
// MI455X (gfx1250) — compile-verified
//
#include <hip/hip_runtime.h>
#include <math.h>

#define NT 100          // number of thresholds in the grid search
#define QMINF (-128.0f)
#define QMAXF (127.0f)

typedef __attribute__((ext_vector_type(2))) float v2f;
typedef __attribute__((ext_vector_type(8))) float v8f;

// ---------------- wave32 reductions ----------------
__device__ inline float wave_sum32(float v) {
#pragma unroll
  for (int o = 16; o > 0; o >>= 1) v += __shfl_xor(v, o, 32);
  return v;
}
__device__ inline float wave_min32(float v) {
#pragma unroll
  for (int o = 16; o > 0; o >>= 1) v = fminf(v, __shfl_xor(v, o, 32));
  return v;
}
__device__ inline float wave_max32(float v) {
#pragma unroll
  for (int o = 16; o > 0; o >>= 1) v = fmaxf(v, __shfl_xor(v, o, 32));
  return v;
}

// ---------------- kernel 1: per-block min/max ----------------
__global__ void minmax_part(const float* __restrict__ x, long long n,
                            float* __restrict__ bmin, float* __restrict__ bmax) {
  long long tid = (long long)blockIdx.x * blockDim.x + threadIdx.x;
  long long stride = (long long)gridDim.x * blockDim.x;
  long long n4 = n >> 2;
  float mn = INFINITY, mx = -INFINITY;
  const float4* __restrict__ x4 = (const float4*)x;
  for (long long i = tid; i < n4; i += stride) {
    float4 v = x4[i];
    mn = fminf(mn, fminf(fminf(v.x, v.y), fminf(v.z, v.w)));
    mx = fmaxf(mx, fmaxf(fmaxf(v.x, v.y), fmaxf(v.z, v.w)));
  }
  for (long long i = (n4 << 2) + tid; i < n; i += stride) {  // scalar tail
    float v = x[i];
    mn = fminf(mn, v);
    mx = fmaxf(mx, v);
  }
  mn = wave_min32(mn);
  mx = wave_max32(mx);
  __shared__ float smn[8], smx[8];
  int w = threadIdx.x >> 5, l = threadIdx.x & 31;
  if (l == 0) { smn[w] = mn; smx[w] = mx; }
  __syncthreads();
  if (threadIdx.x == 0) {
    float a = smn[0], b = smx[0];
    int nw = (blockDim.x + 31) >> 5;
    for (int k = 1; k < nw; ++k) { a = fminf(a, smn[k]); b = fmaxf(b, smx[k]); }
    bmin[blockIdx.x] = a;
    bmax[blockIdx.x] = b;
  }
}

// ---------------- kernel 2: xrange = max(|min|, max) ----------------
__global__ void minmax_final(const float* __restrict__ bmin,
                             const float* __restrict__ bmax, int nb,
                             float* __restrict__ wsf) {
  int t = threadIdx.x;
  float mn = INFINITY, mx = -INFINITY;
  for (int i = t; i < nb; i += blockDim.x) {
    mn = fminf(mn, bmin[i]);
    mx = fmaxf(mx, bmax[i]);
  }
  mn = wave_min32(mn);
  mx = wave_max32(mx);
  __shared__ float smn[8], smx[8];
  int w = t >> 5, l = t & 31;
  if (l == 0) { smn[w] = mn; smx[w] = mx; }
  __syncthreads();
  if (t == 0) {
    float a = smn[0], b = smx[0];
    int nw = (blockDim.x + 31) >> 5;
    for (int k = 1; k < nw; ++k) { a = fminf(a, smn[k]); b = fmaxf(b, smx[k]); }
    wsf[0] = fmaxf(fabsf(a), b);  // jnp.maximum(|x_min|, x_max)
  }
}

// ---------------- kernel 3: per-block per-threshold Lp sums ----------------
// Each lane caches 16 elements in VGPRs, loops all 100 thresholds over them:
// x read from HBM exactly once; kernel is transcendental-bound thereafter.
__global__ void __launch_bounds__(256) score_partial(
    const float* __restrict__ x, long long n, const float* __restrict__ wsf,
    float* __restrict__ parts) {
  __shared__ float s_scale[NT];
  __shared__ float s_inv[NT];
  __shared__ float s_acc[8 * 104];  // per-wave accumulators, padded stride
  const int t = threadIdx.x;
  const float xrange = wsf[0];
  if (t < NT) {
    float thres = xrange * (1.0f / (float)NT) * (float)(t + 1);
    float sc = fmaxf(thres * (1.0f / 127.5f), 1e-8f);  // calculate_qparams
    s_scale[t] = sc;
    s_inv[t] = 1.0f / sc;
  }
  for (int k = t; k < 8 * 104; k += blockDim.x) s_acc[k] = 0.0f;
  __syncthreads();

  const int wv = t >> 5, ln = t & 31;
  const long long tid = (long long)blockIdx.x * blockDim.x + t;
  const long long stride = (long long)gridDim.x * blockDim.x;
  const long long n4t = (n + 3) >> 2;  // float4 slots incl. ragged tail
  const float4* __restrict__ x4 = (const float4*)x;

  for (long long base = tid; base < n4t; base += stride * 4) {
    float e[16];
#pragma unroll
    for (int j = 0; j < 4; ++j) {  // 4 fully-coalesced b128 loads per lane
      long long i4 = base + (long long)j * stride;
      float4 v = make_float4(0.0f, 0.0f, 0.0f, 0.0f);  // zero pad: adds 0 to every score
      if (i4 < n4t) {
        long long b0 = i4 << 2;
        if (b0 + 3 < n) {
          v = x4[i4];
        } else {
          if (b0 < n) v.x = x[b0];
          if (b0 + 1 < n) v.y = x[b0 + 1];
          if (b0 + 2 < n) v.z = x[b0 + 2];
        }
      }
      e[4 * j + 0] = v.x; e[4 * j + 1] = v.y;
      e[4 * j + 2] = v.z; e[4 * j + 3] = v.w;
    }
    long long nb = base + stride * 4;
    if (nb < n4t) __builtin_prefetch(&x4[nb], 0, 0);  // global_prefetch_b8

    for (int i = 0; i < NT; ++i) {
      const float sc = s_scale[i];
      const float inv = s_inv[i];
      float p = 0.0f;
#pragma unroll
      for (int j = 0; j < 16; ++j) {
        float q = rintf(e[j] * inv);            // jnp.round: RNE
        q = fminf(fmaxf(q, QMINF), QMAXF);       // clip to [-128,127]
        float d = fabsf(fmaf(q, sc, -e[j]));     // |x_q - x|
        // d^2.4 = exp2(2.4*log2(d)); native v_log_f32 / v_exp_f32.
        // d==0 -> log2 = -inf -> exp2(-inf) = 0, matching 0^2.4.
        p += __builtin_amdgcn_exp2f(2.4f * __builtin_amdgcn_logf(d));
      }
      p = wave_sum32(p);
      if (ln == 0) s_acc[wv * 104 + i] += p;  // deterministic per-wave slot
    }
  }
  __syncthreads();
  for (int i = t; i < NT; i += blockDim.x) {
    float s = 0.0f;
    int nw = (blockDim.x + 31) >> 5;
    for (int w = 0; w < nw; ++w) s += s_acc[w * 104 + i];
    parts[(long long)blockIdx.x * NT + i] = s;
  }
}

// ---------------- kernel 4: WMMA cross-block reduction + argmin ----------------
// One wave (EXEC all ones). Sum over nbc blocks per threshold via chained
// V_WMMA_F32_16X16X4_F32 with A = ones(16x4): D[:,n] = sum_k B[k][n] + C.
// Summing over all K is invariant to the exact K<->(VGPR,half) mapping.
__global__ void final_select(const float* __restrict__ parts, int nbc,
                             const float* __restrict__ wsf,
                             float* __restrict__ out) {
  __shared__ float s_sums[112];
  const int t = threadIdx.x;
  const int half = t >> 4;
  const int col = t & 15;
  v2f aones;
  aones[0] = 1.0f;
  aones[1] = 1.0f;
  for (int chunk = 0; chunk < 7; ++chunk) {  // 7*16 = 112 >= 100 thresholds
    const int thr = chunk * 16 + col;
    v8f c = {};
    for (int b = 0; b < nbc; b += 4) {
      int b0 = b + half;
      int b1 = b + 2 + half;
      v2f bm;
      bm[0] = (thr < NT && b0 < nbc) ? parts[(long long)b0 * NT + thr] : 0.0f;
      bm[1] = (thr < NT && b1 < nbc) ? parts[(long long)b1 * NT + thr] : 0.0f;
      c = __builtin_amdgcn_wmma_f32_16x16x4_f32(false, aones, false, bm,
                                                (short)0, c, false, false);
    }
    if (t < 16 && thr < NT) s_sums[thr] = c[0];  // row M=0, N=col
  }
  __syncthreads();
  if (t == 0) {
    float xrange = wsf[0];
    float best = INFINITY;
    int bi = 1;
    for (int i = 1; i <= NT; ++i) {  // ascending scan, strict '<' == reference
      float s = s_sums[i - 1];       // argmin over sums == argmin over means
      if (s < best) { best = s; bi = i; }
    }
    float thr = xrange * (1.0f / (float)NT) * (float)bi;
    out[0] = -thr;
    out[1] = thr;
  }
}

// ---------------- host launcher ----------------
extern "C" void kernel_launch(void* const* d_in, const int* in_sizes, int n_in,
                              void* d_out, int out_size, void* d_ws,
                              size_t ws_size, hipStream_t stream) {
  const float* x = (const float*)d_in[0];
  long long n = (long long)in_sizes[0];
  float* wsf = (float*)d_ws;
  long long wsn = (long long)(ws_size / sizeof(float));

  // ws layout: [0..15] scalars | bmin[nba] | bmax[nba] | parts[nbc*NT]
  int nba = 1024;
  while (nba > 1 && 16 + 2LL * nba + NT > wsn) nba >>= 1;
  int nbc = 512;
  while (nbc > 1 && 16 + 2LL * nba + (long long)nbc * NT > wsn) nbc >>= 1;

  float* bmin = wsf + 16;
  float* bmax = bmin + nba;
  float* parts = bmax + nba;

  minmax_part<<<nba, 256, 0, stream>>>(x, n, bmin, bmax);
  minmax_final<<<1, 256, 0, stream>>>(bmin, bmax, nba, wsf);
  score_partial<<<nbc, 256, 0, stream>>>(x, n, wsf, parts);
  final_select<<<1, 32, 0, stream>>>(parts, nbc, wsf, (float*)d_out);
}